// GraphConvolutionLayer_86766929313949
// MI455X (gfx1250) — compile-verified
//
#include <hip/hip_runtime.h>
#include <hip/hip_bf16.h>

#define IN_F  256
#define OUT_F 128

typedef __attribute__((ext_vector_type(2))) float v2f;
typedef __attribute__((ext_vector_type(8))) float v8f;

// ---------------------------------------------------------------------------
// Kernel 1: support = x @ W  via V_WMMA_F32_16X16X4_F32
//   grid.x = N/16 (N = 100000 -> 6250, exact), block = 256 threads = 8 waves.
//   Block computes a 16-row x 128-col output slab; wave w owns cols [16w,16w+16).
//   x tile (16 x 256 f32 = 16 KB) staged in LDS, reused by all 8 waves.
// ---------------------------------------------------------------------------
__global__ __launch_bounds__(256) void gcn_gemm_wmma_f32(
    const float* __restrict__ x, const float* __restrict__ w,
    float* __restrict__ support)
{
    __shared__ float xs[16 * IN_F];   // 16 KB

    const int tid  = threadIdx.x;
    const int wave = tid >> 5;        // 0..7
    const int lane = tid & 31;        // wave32
    const int row0 = blockIdx.x * 16;

    // Cooperative load of the 16x256 x tile: 1024 float4, 4 per thread.
    {
        const float4* xg = reinterpret_cast<const float4*>(x);
        float4*       xl = reinterpret_cast<float4*>(xs);
        #pragma unroll
        for (int i = 0; i < 4; ++i) {
            const int s   = tid * 4 + i;            // 0..1023
            const int r   = s >> 6;                 // row in tile (64 float4/row)
            const int c4  = s & 63;                 // float4 col
            xl[s] = xg[(size_t)(row0 + r) * (IN_F / 4) + c4];
        }
    }
    __syncthreads();

    // WMMA fragment coordinates (ISA 7.12.2, 32-bit A 16x4 / B 4x16 layouts):
    //   A: lane (k/2)*16 + m holds A[m][k] in its VGPR number k%2
    //   B: lane (k/2)*16 + n holds B[k][n] in its VGPR number k%2
    const int mn   = lane & 15;          // m for A, n for B
    const int kq   = (lane >> 4) << 1;   // 0 or 2
    const int col0 = wave * 16;

    v8f acc = {};
    #pragma unroll 8
    for (int k = 0; k < IN_F; k += 4) {
        // A from LDS: two consecutive f32 at even index -> ds_load_b64
        v2f a = *reinterpret_cast<const v2f*>(&xs[mn * IN_F + k + kq]);
        // B from global (W is 128 KB, L1/L2 resident across all blocks)
        v2f b;
        b[0] = w[(size_t)(k + kq    ) * OUT_F + col0 + mn];
        b[1] = w[(size_t)(k + kq + 1) * OUT_F + col0 + mn];
        // 8 args: (neg_a, A, neg_b, B, c_mod, C, reuse_a, reuse_b)
        acc = __builtin_amdgcn_wmma_f32_16x16x4_f32(
            false, a, false, b, (short)0, acc, false, false);
    }

    // D layout: VGPR r -> M = r + 8*(lane>=16), N = lane&15 (coalesced stores)
    const int mrow = (lane >> 4) * 8;
    #pragma unroll
    for (int r = 0; r < 8; ++r) {
        support[(size_t)(row0 + mrow + r) * OUT_F + col0 + mn] = acc[r];
    }
}

// ---------------------------------------------------------------------------
// Kernel 2: out[n, :] = bias[:]   (fold bias into the aggregation init)
// ---------------------------------------------------------------------------
__global__ __launch_bounds__(256) void gcn_bias_init(
    float* __restrict__ out, const float* __restrict__ bias, int total4)
{
    const int i = blockIdx.x * blockDim.x + threadIdx.x;   // one float4 each
    if (i < total4) {
        const int f4 = i & (OUT_F / 4 - 1);                // 32 float4 per row
        reinterpret_cast<float4*>(out)[i] =
            reinterpret_cast<const float4*>(bias)[f4];
    }
}

// ---------------------------------------------------------------------------
// Kernel 3: COO SpMM scatter. One wave per edge, lane handles 4 features.
//   support (51 MB) and out (51 MB) both fit in the 192 MB L2, so the
//   global_atomic_add_f32 RMWs resolve at L2 bandwidth, not HBM.
// ---------------------------------------------------------------------------
__global__ __launch_bounds__(256) void gcn_spmm_scatter(
    const float* __restrict__ support, const float* __restrict__ ew,
    const int* __restrict__ rowi, const int* __restrict__ coli,
    float* __restrict__ out, int nedges)
{
    const int gid  = blockIdx.x * blockDim.x + threadIdx.x;
    const int e    = gid >> 5;
    const int lane = threadIdx.x & 31;
    if (e >= nedges) return;

    const int   c  = coli[e];
    const int   r  = rowi[e];
    const float wg = ew[e];

    const float4 v =
        reinterpret_cast<const float4*>(support + (size_t)c * OUT_F)[lane];

    float* dst = out + (size_t)r * OUT_F + lane * 4;
    atomicAdd(dst + 0, v.x * wg);
    atomicAdd(dst + 1, v.y * wg);
    atomicAdd(dst + 2, v.z * wg);
    atomicAdd(dst + 3, v.w * wg);
}

// ---------------------------------------------------------------------------
extern "C" void kernel_launch(void* const* d_in, const int* in_sizes, int n_in,
                              void* d_out, int out_size, void* d_ws, size_t ws_size,
                              hipStream_t stream)
{
    const float* x    = (const float*)d_in[0];
    const float* w    = (const float*)d_in[1];
    const float* bias = (const float*)d_in[2];
    const float* ew   = (const float*)d_in[3];
    const int*   rowi = (const int*)d_in[4];
    const int*   coli = (const int*)d_in[5];
    float*       out  = (float*)d_out;
    float*       supp = (float*)d_ws;          // N * OUT_F * 4B = 51.2 MB

    const int n_nodes = in_sizes[0] / IN_F;    // 100000 (divisible by 16)
    const int n_edges = in_sizes[3];           // 1600000

    // 1) support = x @ W  (WMMA f32)
    gcn_gemm_wmma_f32<<<n_nodes / 16, 256, 0, stream>>>(x, w, supp);

    // 2) out = broadcast(bias)
    const int total4 = n_nodes * (OUT_F / 4);
    gcn_bias_init<<<(total4 + 255) / 256, 256, 0, stream>>>(out, bias, total4);

    // 3) out[row] += support[col] * edge_weight  (wave-per-edge scatter)
    const long long threads = (long long)n_edges * 32;
    gcn_spmm_scatter<<<(int)((threads + 255) / 256), 256, 0, stream>>>(
        supp, ew, rowi, coli, out, n_edges);
}